// GravnetModel_39161511804938
// MI455X (gfx1250) — compile-verified
//
#include <hip/hip_runtime.h>

#define NTOT 32768
#define BEV  16
#define NPN  2048

typedef float v2f __attribute__((ext_vector_type(2)));
typedef float v8f __attribute__((ext_vector_type(8)));

__device__ __forceinline__ float eluf(float x) { return x > 0.f ? x : (__expf(x) - 1.f); }

// monotonic float <-> sortable uint (ascending uint == ascending float)
// branchless: neg -> ~bits, pos -> bits|0x80000000
__device__ __forceinline__ unsigned f32_to_key(float f) {
  unsigned b = __float_as_uint(f);
  return b ^ (((unsigned)((int)b >> 31)) | 0x80000000u);
}
__device__ __forceinline__ float key_to_f32(unsigned k) {
  return __uint_as_float(k ^ (((unsigned)((int)(~k) >> 31)) | 0x80000000u));
}

// 32-bit LDS byte offset of a generic pointer known to point into LDS
__device__ __forceinline__ unsigned as3_off(const void* p) {
  return (unsigned)(unsigned long long)(__attribute__((address_space(3))) const char*)p;
}

// CDNA5 async copy: global -> LDS, 16B per lane, tracked by ASYNCcnt
__device__ __forceinline__ void async_load_b128(unsigned lds_off, const void* gptr) {
  asm volatile("global_load_async_to_lds_b128 %0, %1, off"
               :: "v"(lds_off), "v"(gptr) : "memory");
}
__device__ __forceinline__ void wait_asynccnt0() {
  asm volatile("s_wait_asynccnt 0" ::: "memory");
}

// ---------------------------------------------------------------------------
// x[N,9] @ W_in[9,6] -> x0[N,6]
__global__ void embed_kernel(const float* __restrict__ x, const float* __restrict__ Win,
                             float* __restrict__ x0) {
  int i = blockIdx.x * blockDim.x + threadIdx.x;
  if (i >= NTOT) return;
  const float* xi = x + (size_t)i * 9;
  float r[6] = {0.f, 0.f, 0.f, 0.f, 0.f, 0.f};
  for (int kk = 0; kk < 9; kk++) {
    float v = xi[kk];
    for (int d = 0; d < 6; d++) r[d] += v * Win[kk * 6 + d];
  }
  for (int d = 0; d < 6; d++) x0[(size_t)i * 6 + d] = r[d];
}

// ---------------------------------------------------------------------------
// s = xin@Ws + bs  [N,3];  h = xin@Wh + bh  [N,12]
__global__ void sh_kernel(const float* __restrict__ xin, int xstride,
                          const float* __restrict__ Ws, const float* __restrict__ bs,
                          const float* __restrict__ Wh, const float* __restrict__ bh,
                          float* __restrict__ s, float* __restrict__ h) {
  int i = blockIdx.x * blockDim.x + threadIdx.x;
  if (i >= NTOT) return;
  const float* xi = xin + (size_t)i * xstride;
  float sv[3] = {bs[0], bs[1], bs[2]};
  float hv[12];
  for (int d = 0; d < 12; d++) hv[d] = bh[d];
  for (int kk = 0; kk < 6; kk++) {
    float v = xi[kk];
    for (int d = 0; d < 3; d++)  sv[d] += v * Ws[kk * 3 + d];
    for (int d = 0; d < 12; d++) hv[d] += v * Wh[kk * 12 + d];
  }
  for (int d = 0; d < 3; d++)  s[(size_t)i * 3 + d]  = sv[d];
  for (int d = 0; d < 12; d++) h[(size_t)i * 12 + d] = hv[d];
}

// ---------------------------------------------------------------------------
// GravNet core: async-stage s/h into LDS -> WMMA gram -> d2 keys in LDS ->
// per-row radix top-k (parallel bin scan) -> fused exp-weighted mean/max agg.
// grid = BEV*128 blocks, 512 threads (16 waves); wave w owns row w for select.
// LDS: keys 128KB | hist 16KB | rowSq 64B | sstage 24KB | hstage 96KB  (~270KB)
__global__ __launch_bounds__(512) void knn_agg_kernel(const float* __restrict__ s,
                                                      const float* __restrict__ h,
                                                      float* __restrict__ agg, int k) {
  extern __shared__ char smem_raw[];
  unsigned* keys  = (unsigned*)smem_raw;         // 16*2048
  unsigned* hist  = keys + 16 * NPN;             // 16*256
  float*    rowSq = (float*)(hist + 16 * 256);   // 16
  float*    sstage = rowSq + 16;                 // 2048*3
  float*    hstage = sstage + NPN * 3;           // 2048*12

  const int tid   = threadIdx.x;
  const int lane  = tid & 31;
  const int wv    = tid >> 5;              // 0..15
  const int event = blockIdx.x >> 7;       // /128
  const int rtile = blockIdx.x & 127;
  const int rowBase = event * NPN + rtile * 16;   // global node index of tile row 0

  // ---- phase 0: async-stage this event's s (24KB) and h (96KB) into LDS ----
  {
    const char* ssrc = (const char*)(s + (size_t)event * NPN * 3);
    const char* hsrc = (const char*)(h + (size_t)event * NPN * 12);
    const unsigned sbase = as3_off(sstage);
    const unsigned hbase = as3_off(hstage);
    for (int off = tid * 16; off < NPN * 3 * 4; off += 512 * 16)
      async_load_b128(sbase + off, ssrc + off);
    for (int off = tid * 16; off < NPN * 12 * 4; off += 512 * 16)
      async_load_b128(hbase + off, hsrc + off);
    wait_asynccnt0();
  }
  __syncthreads();

  if (tid < 16) {
    const float* sr = sstage + (size_t)(rtile * 16 + tid) * 3;
    rowSq[tid] = sr[0] * sr[0] + sr[1] * sr[1] + sr[2] * sr[2];
  }
  __syncthreads();

  // ---- phase 1: gram tiles via V_WMMA_F32_16X16X4_F32 ----
  const int m = lane & 15;
  const float* arow = sstage + (size_t)(rtile * 16 + m) * 3;
  float a0, a1;
  if (lane < 16) { a0 = arow[0]; a1 = arow[1]; }
  else           { a0 = arow[2]; a1 = 0.0f; }
  v2f A; A.x = a0; A.y = a1;
  const int mhalf = (lane >= 16) ? 8 : 0;

  for (int ct = wv; ct < NPN / 16; ct += 16) {
    const float* brow = sstage + (size_t)(ct * 16 + m) * 3;
    float b0, b1;
    if (lane < 16) { b0 = brow[0]; b1 = brow[1]; }
    else           { b0 = brow[2]; b1 = 0.0f; }
    float part  = b0 * b0 + b1 * b1;
    float colSq = __shfl(part, m, 32) + __shfl(part, m + 16, 32);
    v2f Bm; Bm.x = b0; Bm.y = b1;
    v8f acc = {0.f, 0.f, 0.f, 0.f, 0.f, 0.f, 0.f, 0.f};
    acc = __builtin_amdgcn_wmma_f32_16x16x4_f32(false, A, false, Bm, (short)0, acc,
                                                false, false);
#pragma unroll
    for (int v = 0; v < 8; v++) {
      int M = v + mhalf;
      float d2 = rowSq[M] + colSq - 2.0f * acc[v];
      keys[M * NPN + ct * 16 + m] = f32_to_key(d2);
    }
  }
  __syncthreads();

  // ---- phase 2: per-wave radix select on its row (4 x 8-bit passes) ----
  unsigned* krow = keys + wv * NPN;
  unsigned* hrow = hist + wv * 256;
  unsigned prefix = 0, maskSoFar = 0;
  unsigned kRem = (unsigned)k;
  unsigned cLess = 0;
  for (int pass = 0; pass < 4; pass++) {
    int shift = 24 - 8 * pass;
    for (int i = lane; i < 256; i += 32) hrow[i] = 0;   // same-wave LDS, in order
    for (int j = lane; j < NPN; j += 32) {
      unsigned key = krow[j];
      if ((key & maskSoFar) == prefix) atomicAdd(&hrow[(key >> shift) & 255u], 1u);
    }
    // parallel bin scan: lane owns 8 bins, prefix-sum across lanes
    unsigned localCnt[8];
    unsigned localSum = 0;
#pragma unroll
    for (int b = 0; b < 8; b++) { unsigned c = hrow[lane * 8 + b]; localCnt[b] = c; localSum += c; }
    unsigned pre = localSum;
#pragma unroll
    for (int off = 1; off < 32; off <<= 1) {
      unsigned t = (unsigned)__shfl_up((int)pre, off, 32);
      if (lane >= off) pre += t;
    }
    unsigned excl = pre - localSum;   // total in lanes below
    bool hasit = (excl < kRem) && (kRem <= excl + localSum);
    unsigned dsel = 0, cumB = 0;
    if (hasit) {
      unsigned c2 = excl;
      int b = 0;
#pragma unroll
      for (b = 0; b < 8; b++) {
        if (c2 + localCnt[b] >= kRem) break;
        c2 += localCnt[b];
      }
      dsel = (unsigned)(lane * 8 + b);
      cumB = c2;
    }
    unsigned long long bal = __ballot(hasit);
    int srcl = __ffsll(bal) - 1;
    dsel = (unsigned)__shfl((int)dsel, srcl, 32);
    unsigned cum = (unsigned)__shfl((int)cumB, srcl, 32);
    kRem -= cum;
    cLess += cum;
    prefix    |= dsel << shift;
    maskSoFar |= 0xFFu << shift;
  }
  const unsigned T = prefix;                    // k-th smallest key
  const unsigned needed = (unsigned)k - cLess;  // #equal-to-T to take (>=1)

  // ---- phase 3: fused selection + exp-weighted mean/max aggregation ----
  float wsum[12], wmax[12];
#pragma unroll
  for (int d = 0; d < 12; d++) { wsum[d] = 0.f; wmax[d] = -3.402823466e38f; }
  unsigned eqSeen = 0;
  for (int base = 0; base < NPN; base += 32) {
    int j = base + lane;
    unsigned key = krow[j];
    bool eq = (key == T);
    unsigned long long bal = __ballot(eq);
    unsigned prefEq = (unsigned)__popcll(bal & ((1ull << lane) - 1ull));
    bool take = (key < T) || (eq && (eqSeen + prefEq < needed));
    eqSeen += (unsigned)__popcll(bal);
    if (take) {
      float d2  = key_to_f32(key);
      float wgt = __expf(-10.0f * fmaxf(d2, 0.0f));
      const float* hj = hstage + (size_t)j * 12;   // LDS gather
#pragma unroll
      for (int d = 0; d < 12; d++) {
        float mv = wgt * hj[d];
        wsum[d] += mv;
        wmax[d] = fmaxf(wmax[d], mv);
      }
    }
  }
#pragma unroll
  for (int off = 16; off > 0; off >>= 1) {
#pragma unroll
    for (int d = 0; d < 12; d++) {
      wsum[d] += __shfl_down(wsum[d], off, 32);
      wmax[d] = fmaxf(wmax[d], __shfl_down(wmax[d], off, 32));
    }
  }
  if (lane == 0) {
    float inv = 1.0f / (float)k;
    float* ao = agg + (size_t)(rowBase + wv) * 24;
    for (int d = 0; d < 12; d++) { ao[d] = wsum[d] * inv; ao[12 + d] = wmax[d]; }
  }
}

// ---------------------------------------------------------------------------
// out = xin@Wg1 + agg@Wg2 + bg2 [24]; t27=cat(out,s); p=elu(t27@Wp1+bp1);
// t = elu(p@Wp2+bp2) [6]
__global__ void post_kernel(const float* __restrict__ xin, int xstride,
                            const float* __restrict__ s, const float* __restrict__ agg,
                            const float* __restrict__ Wg1, const float* __restrict__ Wg2,
                            const float* __restrict__ bg2,
                            const float* __restrict__ Wp1, const float* __restrict__ bp1,
                            const float* __restrict__ Wp2, const float* __restrict__ bp2,
                            float* __restrict__ t) {
  int i = blockIdx.x * blockDim.x + threadIdx.x;
  if (i >= NTOT) return;
  const float* xi = xin + (size_t)i * xstride;
  float out[24];
  for (int c = 0; c < 24; c++) out[c] = bg2[c];
  for (int kk = 0; kk < 6; kk++) {
    float v = xi[kk];
    for (int c = 0; c < 24; c++) out[c] += v * Wg1[kk * 24 + c];
  }
  const float* ai = agg + (size_t)i * 24;
  for (int kk = 0; kk < 24; kk++) {
    float v = ai[kk];
    for (int c = 0; c < 24; c++) out[c] += v * Wg2[kk * 24 + c];
  }
  float p[6];
  for (int d = 0; d < 6; d++) p[d] = bp1[d];
  for (int kk = 0; kk < 24; kk++) {
    float v = out[kk];
    for (int d = 0; d < 6; d++) p[d] += v * Wp1[kk * 6 + d];
  }
  for (int kk = 0; kk < 3; kk++) {
    float v = s[(size_t)i * 3 + kk];
    for (int d = 0; d < 6; d++) p[d] += v * Wp1[(24 + kk) * 6 + d];
  }
  for (int d = 0; d < 6; d++) p[d] = eluf(p[d]);
  float t2[6];
  for (int d = 0; d < 6; d++) t2[d] = bp2[d];
  for (int kk = 0; kk < 6; kk++) {
    float v = p[kk];
    for (int d = 0; d < 6; d++) t2[d] += v * Wp2[kk * 6 + d];
  }
  for (int d = 0; d < 6; d++) t[(size_t)i * 6 + d] = eluf(t2[d]);
}

// ---------------------------------------------------------------------------
// per-event mean/min/max of t[6] + node feats -> xout = elu(ex@Wex+bex)
// grid = BEV blocks, 256 threads
__global__ void exch_kernel(const float* __restrict__ t, const float* __restrict__ Wex,
                            const float* __restrict__ bex, float* __restrict__ xout,
                            int xstride) {
  __shared__ float rsum[256 * 6];
  __shared__ float rmin[256 * 6];
  __shared__ float rmax[256 * 6];
  __shared__ float stats[18];
  int e = blockIdx.x, tid = threadIdx.x;
  const float* te = t + (size_t)e * NPN * 6;
  float ls[6], lmn[6], lmx[6];
  for (int d = 0; d < 6; d++) { ls[d] = 0.f; lmn[d] = 3.402823466e38f; lmx[d] = -3.402823466e38f; }
  for (int j = tid; j < NPN; j += 256) {
    for (int d = 0; d < 6; d++) {
      float v = te[(size_t)j * 6 + d];
      ls[d] += v; lmn[d] = fminf(lmn[d], v); lmx[d] = fmaxf(lmx[d], v);
    }
  }
  for (int d = 0; d < 6; d++) { rsum[tid * 6 + d] = ls[d]; rmin[tid * 6 + d] = lmn[d]; rmax[tid * 6 + d] = lmx[d]; }
  __syncthreads();
  for (int off = 128; off > 0; off >>= 1) {
    if (tid < off) {
      for (int d = 0; d < 6; d++) {
        rsum[tid * 6 + d] += rsum[(tid + off) * 6 + d];
        rmin[tid * 6 + d] = fminf(rmin[tid * 6 + d], rmin[(tid + off) * 6 + d]);
        rmax[tid * 6 + d] = fmaxf(rmax[tid * 6 + d], rmax[(tid + off) * 6 + d]);
      }
    }
    __syncthreads();
  }
  if (tid < 6) {
    stats[tid]      = rsum[tid] * (1.0f / (float)NPN);
    stats[6 + tid]  = rmin[tid];
    stats[12 + tid] = rmax[tid];
  }
  __syncthreads();
  for (int j = tid; j < NPN; j += 256) {
    float ex[24];
    for (int d = 0; d < 18; d++) ex[d] = stats[d];
    for (int d = 0; d < 6; d++)  ex[18 + d] = te[(size_t)j * 6 + d];
    float o[6];
    for (int d = 0; d < 6; d++) o[d] = bex[d];
    for (int kk = 0; kk < 24; kk++) {
      float v = ex[kk];
      for (int d = 0; d < 6; d++) o[d] += v * Wex[kk * 6 + d];
    }
    size_t node = (size_t)e * NPN + j;
    for (int d = 0; d < 6; d++) xout[node * xstride + d] = eluf(o[d]);
  }
}

// ---------------------------------------------------------------------------
// y24 -> elu(Wd0) -> 3x elu(Wd) -> elu(Wh1) -> h1[N,64]
__global__ void head1_kernel(const float* __restrict__ feats,
                             const float* __restrict__ Wd0, const float* __restrict__ bd0,
                             const float* __restrict__ Wd,  const float* __restrict__ bd,
                             const float* __restrict__ Wh1, const float* __restrict__ bh1,
                             float* __restrict__ h1) {
  int i = blockIdx.x * blockDim.x + threadIdx.x;
  if (i >= NTOT) return;
  const float* f = feats + (size_t)i * 24;
  float y[6];
  for (int d = 0; d < 6; d++) y[d] = bd0[d];
  for (int kk = 0; kk < 24; kk++) {
    float v = f[kk];
    for (int d = 0; d < 6; d++) y[d] += v * Wd0[kk * 6 + d];
  }
  for (int d = 0; d < 6; d++) y[d] = eluf(y[d]);
  for (int j = 0; j < 3; j++) {
    float z[6];
    for (int d = 0; d < 6; d++) z[d] = bd[j * 6 + d];
    for (int kk = 0; kk < 6; kk++) {
      float v = y[kk];
      for (int d = 0; d < 6; d++) z[d] += v * Wd[j * 36 + kk * 6 + d];
    }
    for (int d = 0; d < 6; d++) y[d] = eluf(z[d]);
  }
  float* o = h1 + (size_t)i * 64;
  for (int d = 0; d < 64; d++) {
    float acc = bh1[d];
    for (int kk = 0; kk < 6; kk++) acc += y[kk] * Wh1[kk * 64 + d];
    o[d] = eluf(acc);
  }
}

// ---------------------------------------------------------------------------
// h2 = elu(h1@Wh2+bh2); h3 = h2@Wh3+bh3; out = [h3@Wc+bc, h3@Wb+bb]  (WMMA f32)
// grid = NTOT/128 blocks, 256 threads (8 waves), 1 wave = one 16-row tile.
__global__ __launch_bounds__(256) void head2_kernel(const float* __restrict__ h1,
    const float* __restrict__ Wh2, const float* __restrict__ bh2,
    const float* __restrict__ Wh3, const float* __restrict__ bh3,
    const float* __restrict__ Wc,  const float* __restrict__ bc,
    const float* __restrict__ Wb,  const float* __restrict__ bb,
    float* __restrict__ out) {
  extern __shared__ char smem_raw[];
  float* w2 = (float*)smem_raw;       // 64*64
  float* w3 = w2 + 4096;              // 64*64
  float* b2 = w3 + 4096;              // 64
  float* b3 = b2 + 64;                // 64
  float* trans = b3 + 64;             // 8 waves * 16*64

  const int tid = threadIdx.x, lane = tid & 31, wv = tid >> 5;
  {
    const unsigned w2off = as3_off(w2);
    const unsigned w3off = as3_off(w3);
    for (int off = tid * 16; off < 64 * 64 * 4; off += 256 * 16) {
      async_load_b128(w2off + off, (const char*)Wh2 + off);
      async_load_b128(w3off + off, (const char*)Wh3 + off);
    }
    wait_asynccnt0();
  }
  if (tid < 64) { b2[tid] = bh2[tid]; b3[tid] = bh3[tid]; }
  __syncthreads();

  const int rowBase = (blockIdx.x * 8 + wv) * 16;
  const int n = lane & 15;
  const int klo = (lane < 16) ? 0 : 2;
  const int mhalf = (lane >= 16) ? 8 : 0;
  float* tb = trans + wv * 1024;

  // layer 2: h2 = elu(h1 @ Wh2 + bh2)
  v8f acc[4];
#pragma unroll
  for (int nt = 0; nt < 4; nt++) {
    float bv = b2[nt * 16 + n];
#pragma unroll
    for (int v = 0; v < 8; v++) acc[nt][v] = bv;
  }
  for (int kk = 0; kk < 16; kk++) {
    const float* arow = h1 + (size_t)(rowBase + n) * 64 + 4 * kk + klo;
    v2f A; A.x = arow[0]; A.y = arow[1];
#pragma unroll
    for (int nt = 0; nt < 4; nt++) {
      v2f Bv;
      Bv.x = w2[(4 * kk + klo) * 64 + nt * 16 + n];
      Bv.y = w2[(4 * kk + klo + 1) * 64 + nt * 16 + n];
      acc[nt] = __builtin_amdgcn_wmma_f32_16x16x4_f32(false, A, false, Bv, (short)0,
                                                      acc[nt], false, false);
    }
  }
#pragma unroll
  for (int nt = 0; nt < 4; nt++)
#pragma unroll
    for (int v = 0; v < 8; v++) tb[(v + mhalf) * 64 + nt * 16 + n] = eluf(acc[nt][v]);

  // layer 3: h3 = h2 @ Wh3 + bh3  (A read back from LDS transpose buffer)
  v8f acc3[4];
#pragma unroll
  for (int nt = 0; nt < 4; nt++) {
    float bv = b3[nt * 16 + n];
#pragma unroll
    for (int v = 0; v < 8; v++) acc3[nt][v] = bv;
  }
  for (int kk = 0; kk < 16; kk++) {
    v2f A; A.x = tb[n * 64 + 4 * kk + klo]; A.y = tb[n * 64 + 4 * kk + klo + 1];
#pragma unroll
    for (int nt = 0; nt < 4; nt++) {
      v2f Bv;
      Bv.x = w3[(4 * kk + klo) * 64 + nt * 16 + n];
      Bv.y = w3[(4 * kk + klo + 1) * 64 + nt * 16 + n];
      acc3[nt] = __builtin_amdgcn_wmma_f32_16x16x4_f32(false, A, false, Bv, (short)0,
                                                       acc3[nt], false, false);
    }
  }
#pragma unroll
  for (int nt = 0; nt < 4; nt++)
#pragma unroll
    for (int v = 0; v < 8; v++) tb[(v + mhalf) * 64 + nt * 16 + n] = acc3[nt][v];

  // final: 31 output columns per row (cols 0..29 = Wc, col 30 = Wb)
  if (lane < 31) {
    for (int r = 0; r < 16; r++) {
      const float* hr = tb + r * 64;
      float accv;
      if (lane < 30) {
        accv = bc[lane];
        for (int kk = 0; kk < 64; kk++) accv += hr[kk] * Wc[kk * 30 + lane];
      } else {
        accv = bb[0];
        for (int kk = 0; kk < 64; kk++) accv += hr[kk] * Wb[kk];
      }
      out[(size_t)(rowBase + r) * 31 + lane] = accv;
    }
  }
}

// ---------------------------------------------------------------------------
extern "C" void kernel_launch(void* const* d_in, const int* in_sizes, int n_in,
                              void* d_out, int out_size, void* d_ws, size_t ws_size,
                              hipStream_t stream) {
  const float* x   = (const float*)d_in[0];
  const float* Win = (const float*)d_in[1];
  const float* Ws  = (const float*)d_in[2];
  const float* bs  = (const float*)d_in[3];
  const float* Wh  = (const float*)d_in[4];
  const float* bh  = (const float*)d_in[5];
  const float* Wg1 = (const float*)d_in[6];
  const float* Wg2 = (const float*)d_in[7];
  const float* bg2 = (const float*)d_in[8];
  const float* Wp1 = (const float*)d_in[9];
  const float* bp1 = (const float*)d_in[10];
  const float* Wp2 = (const float*)d_in[11];
  const float* bp2 = (const float*)d_in[12];
  const float* Wex = (const float*)d_in[13];
  const float* bex = (const float*)d_in[14];
  const float* Wd0 = (const float*)d_in[15];
  const float* bd0 = (const float*)d_in[16];
  const float* Wd  = (const float*)d_in[17];
  const float* bd  = (const float*)d_in[18];
  const float* Wh1 = (const float*)d_in[19];
  const float* bh1 = (const float*)d_in[20];
  const float* Wh2 = (const float*)d_in[21];
  const float* bh2 = (const float*)d_in[22];
  const float* Wh3 = (const float*)d_in[23];
  const float* bh3 = (const float*)d_in[24];
  const float* Wc  = (const float*)d_in[25];
  const float* bc  = (const float*)d_in[26];
  const float* Wb  = (const float*)d_in[27];
  const float* bb  = (const float*)d_in[28];
  float* out = (float*)d_out;

  float* ws    = (float*)d_ws;
  float* x0    = ws;                       // N*6
  float* sbuf  = x0    + (size_t)NTOT * 6; // N*3
  float* hbuf  = sbuf  + (size_t)NTOT * 3; // N*12
  float* abuf  = hbuf  + (size_t)NTOT * 12;// N*24
  float* tbuf  = abuf  + (size_t)NTOT * 24;// N*6
  float* feats = tbuf  + (size_t)NTOT * 6; // N*24
  float* h1    = feats + (size_t)NTOT * 24;// N*64

  const int ks[4] = {16, 128, 16, 256};
  dim3 pgrid(NTOT / 256), pblk(256);

  embed_kernel<<<pgrid, pblk, 0, stream>>>(x, Win, x0);

  // LDS: keys + hist + rowSq + sstage + hstage
  const size_t knnlds = (size_t)(16 * NPN + 16 * 256 + 16) * sizeof(unsigned)
                      + (size_t)(NPN * 3 + NPN * 12) * sizeof(float);
  for (int i = 0; i < 4; i++) {
    const float* xin = (i == 0) ? x0 : (feats + (size_t)(i - 1) * 6);
    int xstride = (i == 0) ? 6 : 24;
    sh_kernel<<<pgrid, pblk, 0, stream>>>(xin, xstride, Ws + i * 18, bs + i * 3,
                                          Wh + i * 72, bh + i * 12, sbuf, hbuf);
    knn_agg_kernel<<<dim3(BEV * (NPN / 16)), dim3(512), knnlds, stream>>>(sbuf, hbuf,
                                                                          abuf, ks[i]);
    post_kernel<<<pgrid, pblk, 0, stream>>>(xin, xstride, sbuf, abuf,
        Wg1 + i * 6 * 24, Wg2 + i * 24 * 24, bg2 + i * 24,
        Wp1 + i * 27 * 6, bp1 + i * 6, Wp2 + i * 36, bp2 + i * 6, tbuf);
    exch_kernel<<<dim3(BEV), dim3(256), 0, stream>>>(tbuf, Wex + i * 24 * 6,
                                                     bex + i * 6, feats + (size_t)i * 6, 24);
  }

  head1_kernel<<<pgrid, pblk, 0, stream>>>(feats, Wd0, bd0, Wd, bd, Wh1, bh1, h1);
  const size_t h2lds = (size_t)(4096 + 4096 + 64 + 64 + 8 * 1024) * sizeof(float);
  head2_kernel<<<dim3(NTOT / 128), dim3(256), h2lds, stream>>>(h1, Wh2, bh2, Wh3, bh3,
                                                               Wc, bc, Wb, bb, out);
}